// GCN_72748156059703
// MI455X (gfx1250) — compile-verified
//
#include <hip/hip_runtime.h>

#define NNODES   50000
#define NEDGES   800000
#define FDIM     128
#define NGRAPHS  128
#define NCLASSES 12

typedef __attribute__((ext_vector_type(2))) float v2f;
typedef __attribute__((ext_vector_type(8))) float v8f;

// ---------------------------------------------------------------- utilities
__global__ void gcn_fill_kernel(float* __restrict__ p, float v, int n) {
    int i = blockIdx.x * blockDim.x + threadIdx.x;
    if (i < n) p[i] = v;
}

// deg[c] += 1 per incoming edge (deg buffer pre-filled with 1.0 for self loop)
__global__ void gcn_deg_kernel(const int* __restrict__ col, float* __restrict__ deg, int E) {
    int e = blockIdx.x * blockDim.x + threadIdx.x;
    if (e < E) atomicAdd(&deg[col[e]], 1.0f);
}

// in-place deg -> rsqrt(deg)   (deg >= 1 always, so no zero guard needed)
__global__ void gcn_rsqrt_kernel(float* __restrict__ d, int n) {
    int i = blockIdx.x * blockDim.x + threadIdx.x;
    if (i < n) d[i] = rsqrtf(d[i]);
}

__global__ void gcn_norm_kernel(const int* __restrict__ row, const int* __restrict__ col,
                                const float* __restrict__ dis, float* __restrict__ norm, int E) {
    int e = blockIdx.x * blockDim.x + threadIdx.x;
    if (e < E) norm[e] = dis[row[e]] * dis[col[e]];
}

// ---------------------------------------------------------------- WMMA GEMM
// Y[nrows,128] = X[nrows,128] @ W[128,128], fp32 via V_WMMA_F32_16X16X4_F32.
// block = 256 threads (8 waves); each wave owns one 16x16 output tile of a
// 16-row strip: colBase = waveId*16 covers all 128 output columns.
__global__ void gcn_gemm_kernel(const float* __restrict__ X, const float* __restrict__ W,
                                float* __restrict__ Y) {
    const int wave = threadIdx.x >> 5;
    const int lane = threadIdx.x & 31;
    const int r    = lane & 15;      // M (for A/D) or N (for B/D)
    const int hi   = lane >> 4;      // half-wave select
    const int rowBase = blockIdx.x * 16;
    const int colBase = wave * 16;

    const float* __restrict__ xrow = X + (size_t)(rowBase + r) * FDIM;
    v8f acc = {};

#pragma unroll
    for (int k0 = 0; k0 < FDIM; k0 += 4) {
        // A 16x4: lane (r,hi), VGPR v -> element (M=r, K=k0+2*hi+v)
        v2f a = *(const v2f*)(xrow + k0 + 2 * hi);
        // B 4x16: lane (r,hi), VGPR v -> element (K=k0+2*hi+v, N=colBase+r)
        v2f b;
        b[0] = W[(size_t)(k0 + 2 * hi + 0) * FDIM + colBase + r];
        b[1] = W[(size_t)(k0 + 2 * hi + 1) * FDIM + colBase + r];
        acc = __builtin_amdgcn_wmma_f32_16x16x4_f32(
            /*neg_a=*/false, a, /*neg_b=*/false, b,
            /*c_mod=*/(short)0, acc, /*reuse_a=*/false, /*reuse_b=*/false);
    }

    // D 16x16: VGPR v -> (M = v + 8*hi, N = colBase + r)
#pragma unroll
    for (int v = 0; v < 8; ++v)
        Y[(size_t)(rowBase + v + 8 * hi) * FDIM + colBase + r] = acc[v];
}

// ---------------------------------------------------------------- edge scatter
// one wave per edge: gather xw[row] (float4/lane), scale by norm, atomic-add
// into agg[col]. This is the HBM-bound part of the whole workload.
__global__ void gcn_scatter_kernel(const float* __restrict__ xw, const int* __restrict__ row,
                                   const int* __restrict__ col, const float* __restrict__ norm,
                                   float* __restrict__ agg, int E) {
    int e    = blockIdx.x * (blockDim.x >> 5) + (threadIdx.x >> 5);
    int lane = threadIdx.x & 31;
    if (e >= E) return;
    int   src = row[e];
    int   dst = col[e];
    float nm  = norm[e];
    float4 m = *(const float4*)(xw + (size_t)src * FDIM + lane * 4);
    float* d = agg + (size_t)dst * FDIM + lane * 4;
    atomicAdd(d + 0, m.x * nm);
    atomicAdd(d + 1, m.y * nm);
    atomicAdd(d + 2, m.z * nm);
    atomicAdd(d + 3, m.w * nm);
}

// h_out = relu(agg + xw * dis^2 (self loop) + bias)
__global__ void gcn_epilogue_kernel(const float* __restrict__ agg, const float* __restrict__ xw,
                                    const float* __restrict__ dis, const float* __restrict__ bias,
                                    float* __restrict__ hout, int n) {
    int idx = blockIdx.x * blockDim.x + threadIdx.x;
    if (idx >= n) return;
    int i = idx >> 7;
    int k = idx & 127;
    float d = dis[i];
    float v = agg[idx] + xw[idx] * d * d + bias[k];
    hout[idx] = v > 0.0f ? v : 0.0f;
}

// ---------------------------------------------------------------- pooling + head
__global__ void gcn_pool_kernel(const float* __restrict__ h, const int* __restrict__ batch,
                                float* __restrict__ pooled, int n) {
    int idx = blockIdx.x * blockDim.x + threadIdx.x;
    if (idx >= n) return;
    int i = idx >> 7;
    int k = idx & 127;
    atomicAdd(&pooled[(size_t)batch[i] * FDIM + k], h[idx]);
}

__global__ void gcn_cnt_kernel(const int* __restrict__ batch, float* __restrict__ cnt, int n) {
    int i = blockIdx.x * blockDim.x + threadIdx.x;
    if (i < n) atomicAdd(&cnt[batch[i]], 1.0f);
}

// out[g,c] = (pooled[g,:]/max(cnt,1)) . lin_w[:,c] + lin_b[c]   (128x12, tiny)
__global__ void gcn_head_kernel(const float* __restrict__ pooled, const float* __restrict__ cnt,
                                const float* __restrict__ lw, const float* __restrict__ lb,
                                float* __restrict__ out) {
    int idx = blockIdx.x * blockDim.x + threadIdx.x;
    if (idx >= NGRAPHS * NCLASSES) return;
    int g = idx / NCLASSES;
    int c = idx % NCLASSES;
    float inv = 1.0f / fmaxf(cnt[g], 1.0f);
    float s = 0.0f;
#pragma unroll 8
    for (int k = 0; k < FDIM; ++k)
        s += pooled[(size_t)g * FDIM + k] * lw[k * NCLASSES + c];
    out[idx] = s * inv + lb[c];
}

// ---------------------------------------------------------------- launcher
extern "C" void kernel_launch(void* const* d_in, const int* in_sizes, int n_in,
                              void* d_out, int out_size, void* d_ws, size_t ws_size,
                              hipStream_t stream) {
    const float* x     = (const float*)d_in[0];
    const int*   erow  = (const int*)d_in[1];            // edge_index[0]
    const int*   ecol  = (const int*)d_in[1] + NEDGES;   // edge_index[1]
    const int*   batch = (const int*)d_in[2];
    const float* W[3]  = {(const float*)d_in[3], (const float*)d_in[5], (const float*)d_in[7]};
    const float* B[3]  = {(const float*)d_in[4], (const float*)d_in[6], (const float*)d_in[8]};
    const float* lw    = (const float*)d_in[9];
    const float* lb    = (const float*)d_in[10];
    float*       out   = (float*)d_out;

    // carve workspace (256B aligned slices)
    char* ws = (char*)d_ws;
    auto carve = [&](size_t bytes) -> void* {
        void* p = ws;
        ws += (bytes + 255) & ~(size_t)255;
        return p;
    };
    float* dis    = (float*)carve((size_t)NNODES * 4);
    float* norm   = (float*)carve((size_t)NEDGES * 4);
    float* hbuf   = (float*)carve((size_t)NNODES * FDIM * 4);
    float* xw     = (float*)carve((size_t)NNODES * FDIM * 4);
    float* agg    = (float*)carve((size_t)NNODES * FDIM * 4);
    float* pooled = (float*)carve((size_t)NGRAPHS * FDIM * 4);
    float* cnt    = (float*)carve((size_t)NGRAPHS * 4);

    const int NELEM = NNODES * FDIM;  // 6,400,000

    // 1) symmetric normalization: deg -> dis -> per-edge norm
    gcn_fill_kernel<<<(NNODES + 255) / 256, 256, 0, stream>>>(dis, 1.0f, NNODES);
    gcn_deg_kernel<<<(NEDGES + 255) / 256, 256, 0, stream>>>(ecol, dis, NEDGES);
    gcn_rsqrt_kernel<<<(NNODES + 255) / 256, 256, 0, stream>>>(dis, NNODES);
    gcn_norm_kernel<<<(NEDGES + 255) / 256, 256, 0, stream>>>(erow, ecol, dis, norm, NEDGES);

    // 2) three GCN layers
    const float* hin = x;
    for (int l = 0; l < 3; ++l) {
        gcn_gemm_kernel<<<NNODES / 16, 256, 0, stream>>>(hin, W[l], xw);
        gcn_fill_kernel<<<(NELEM + 255) / 256, 256, 0, stream>>>(agg, 0.0f, NELEM);
        gcn_scatter_kernel<<<(NEDGES + 7) / 8, 256, 0, stream>>>(xw, erow, ecol, norm, agg, NEDGES);
        gcn_epilogue_kernel<<<(NELEM + 255) / 256, 256, 0, stream>>>(agg, xw, dis, B[l], hbuf, NELEM);
        hin = hbuf;
    }

    // 3) global mean pool + linear head
    gcn_fill_kernel<<<(NGRAPHS * FDIM + 255) / 256, 256, 0, stream>>>(pooled, 0.0f, NGRAPHS * FDIM);
    gcn_fill_kernel<<<1, 256, 0, stream>>>(cnt, 0.0f, NGRAPHS);
    gcn_pool_kernel<<<(NELEM + 255) / 256, 256, 0, stream>>>(hbuf, batch, pooled, NELEM);
    gcn_cnt_kernel<<<(NNODES + 255) / 256, 256, 0, stream>>>(batch, cnt, NNODES);
    gcn_head_kernel<<<(NGRAPHS * NCLASSES + 255) / 256, 256, 0, stream>>>(pooled, cnt, lw, lb, out);
}